// SpectralConv2dRealFreq_84550726189372
// MI455X (gfx1250) — compile-verified
//
#include <hip/hip_runtime.h>
#include <stdint.h>

// ---------------- WMMA f32 16x16x4 helpers (wave32) ----------------
typedef float v2f __attribute__((ext_vector_type(2)));
typedef float v8f __attribute__((ext_vector_type(8)));
typedef uint32_t v4u __attribute__((ext_vector_type(4)));
typedef uint32_t v8u __attribute__((ext_vector_type(8)));

__device__ __forceinline__ v8f wmma4(v2f a, v2f b, v8f c) {
  // D = A(16x4) * B(4x16) + C(16x16), fp32
  return __builtin_amdgcn_wmma_f32_16x16x4_f32(
      /*neg_a=*/false, a, /*neg_b=*/false, b,
      /*c_mod=*/(short)0, c, /*reuse_a=*/false, /*reuse_b=*/false);
}

// ---------------- CDNA5 async-to-LDS helpers ----------------
// Flat pointers to LDS carry the LDS byte offset in bits [31:0] (aperture in [63:32]).
__device__ __forceinline__ uint32_t lds_off(const void* p) {
  return (uint32_t)(uintptr_t)p;
}
__device__ __forceinline__ void async_ld_b128(uint32_t ldsaddr, const void* g) {
  asm volatile("global_load_async_to_lds_b128 %0, %1, off"
               :: "v"(ldsaddr), "v"(g) : "memory");
}
__device__ __forceinline__ void async_ld_b32(uint32_t ldsaddr, const void* g) {
  asm volatile("global_load_async_to_lds_b32 %0, %1, off"
               :: "v"(ldsaddr), "v"(g) : "memory");
}
__device__ __forceinline__ void wait_async0() {
  asm volatile("s_wait_asynccnt 0" ::: "memory");
}

// ---------------- CDNA5 Tensor Data Mover: contiguous f32 copy to LDS ----------------
// Builds a D# (CDNA5 ISA): group0 = {count, lds_addr, global_addr, type=2},
// group1 = {data_size=4B, tensor_dim0=tile_dim0=nelem, tensor_dim1=tile_dim1=1, stride}.
// Descriptor lives in SGPRs; tensor ops ignore EXEC, so issue from one wave only.
__device__ __forceinline__ void tdm_load_f32_1d(uint32_t lds_base, const float* g,
                                                uint32_t nelem /* <= 65535 */) {
  uint64_t ga = (uint64_t)(uintptr_t)g;
  v4u g0;
  g0[0] = 1u;                                                  // count=1, user descriptor
  g0[1] = lds_base;                                            // lds_addr (bytes)
  g0[2] = (uint32_t)ga;                                        // global_addr[31:0]
  g0[3] = (uint32_t)((ga >> 32) & 0x01FFFFFFu) | 0x80000000u;  // addr[56:32] | type=2
  v8u g1;
  g1[0] = 0x20000u;                     // data_size=2 -> 4-byte elements
  g1[1] = (nelem & 0xFFFFu) << 16;      // tensor_dim0[15:0]  (bits 63:48)
  g1[2] = (nelem >> 16) | (1u << 16);   // tensor_dim0[31:16] | tensor_dim1=1
  g1[3] = (nelem & 0xFFFFu) << 16;      // tile_dim0          (bits 127:112)
  g1[4] = 1u;                           // tile_dim1=1, tile_dim2=0
  g1[5] = nelem;                        // tensor_dim0_stride[31:0]
  g1[6] = 0u;                           // stride hi / dim1_stride lo
  g1[7] = 0u;
  asm volatile("tensor_load_to_lds %0, %1" :: "s"(g0), "s"(g1) : "memory");
}

// ---------------- workspace layout (float offsets) ----------------
static constexpr size_t TAB_C  = 0;
static constexpr size_t TAB_S  = 6144;
static constexpr size_t TAB_CT = 12288;
static constexpr size_t TAB_ST = 18432;
static constexpr size_t X_RE = 24576;
static constexpr size_t X_IM = X_RE + 1089 * 1024;
static constexpr size_t Y_RE = X_IM + 1089 * 1024;
static constexpr size_t Y_IM = Y_RE + 1089 * 1024;
static constexpr size_t RT_RE = Y_IM + 1089 * 1024;
static constexpr size_t RT_IM = RT_RE + 1089 * 4096;

// ---------------- kernel 0: phase tables ----------------
__global__ void k_tables(float* __restrict__ ws) {
  int t = blockIdx.x * 256 + threadIdx.x;
  if (t >= 48 * 128) return;
  int l = t >> 7, w = t & 127;
  float c = 0.f, s = 0.f;
  if (l < 33) {
    int kv = (l <= 16) ? l : l - 33;
    float ang = 6.28318530717958647692f * (float)(kv * w) / 128.0f;
    c = cosf(ang);
    s = sinf(ang);
  }
  ws[TAB_C + l * 128 + w] = c;
  ws[TAB_S + l * 128 + w] = s;
  ws[TAB_CT + w * 48 + l] = c;
  ws[TAB_ST + w * 48 + l] = s;
}

// ---------------- kernel 1: transpose weights [4096,1089] -> [1089,4096] ----------------
__global__ void k_transpose(const float* __restrict__ wr,
                            const float* __restrict__ wi,
                            float* __restrict__ ws) {
  __shared__ float tile[32][33];
  const float* src = blockIdx.z ? wi : wr;
  float* dst = ws + (blockIdx.z ? RT_IM : RT_RE);
  int m0 = blockIdx.x * 32;
  int c0 = blockIdx.y * 32;
#pragma unroll
  for (int i = 0; i < 4; ++i) {
    int m = m0 + threadIdx.x;
    int co = c0 + threadIdx.y + i * 8;
    if (m < 1089) tile[threadIdx.y + i * 8][threadIdx.x] = src[(size_t)co * 1089 + m];
  }
  __syncthreads();
#pragma unroll
  for (int i = 0; i < 4; ++i) {
    int m = m0 + threadIdx.y + i * 8;
    int co = c0 + threadIdx.x;
    if (m < 1089) dst[(size_t)m * 4096 + co] = tile[threadIdx.x][threadIdx.y + i * 8];
  }
}

// ---------------- kernel 2: forward truncated 2D DFT, one (b,c) image per block ----------------
__global__ void k_fwd(const float* __restrict__ x, float* __restrict__ ws) {
  __shared__ float sX[128 * 128];     // 64 KB image tile (TDM destination)
  __shared__ float sTre[128 * 48];
  __shared__ float sTim[128 * 48];
  const int tid = threadIdx.x;
  const int wv = __builtin_amdgcn_readfirstlane(tid >> 5);  // wave-uniform job id
  const int lane = tid & 31;
  const int half = lane >> 4;
  const int r = lane & 15;
  const int bid = blockIdx.x;  // b*64+c
  const float* xb = x + (size_t)bid * 16384;
  const float* tabC  = ws + TAB_C;
  const float* tabS  = ws + TAB_S;
  const float* tabCt = ws + TAB_CT;
  const float* tabSt = ws + TAB_ST;

  // TDM: one descriptor-driven DMA pulls the whole 64 KB image into LDS.
  if (wv == 0) {
    tdm_load_f32_1d(__builtin_amdgcn_readfirstlane(lds_off(sX)), xb, 16384u);
    __builtin_amdgcn_s_wait_tensorcnt(0);
  }
  __syncthreads();

  // phase 1: T (M=128 h, K=128 w, N=48 l), jobs = 8 ht * 3 lt * 2 comp = 48
  for (int j = wv; j < 48; j += 8) {
    int comp = j & 1;          // 0: re, 1: im
    int tile = j >> 1;
    int ht = tile / 3, lt = tile % 3;
    const float* Bt = comp ? tabSt : tabCt;
    const float* arow = sX + (ht * 16 + r) * 128;
    v8f acc = {};
    for (int k0 = 0; k0 < 128; k0 += 4) {
      int kk = k0 + 2 * half;
      v2f a; a.x = arow[kk]; a.y = arow[kk + 1];
      v2f b; b.x = Bt[kk * 48 + lt * 16 + r]; b.y = Bt[(kk + 1) * 48 + lt * 16 + r];
      acc = wmma4(a, b, acc);
    }
    float* dstT = comp ? sTim : sTre;
    float sgn = comp ? -1.f : 1.f;   // Tim = -x@S
#pragma unroll
    for (int i = 0; i < 8; ++i)
      dstT[(ht * 16 + i + 8 * half) * 48 + lt * 16 + r] = sgn * acc[i];
  }
  __syncthreads();

  // phase 2: X (M=48 k, K=128 h, N=48 l), jobs = 3*3*2 = 18
  float* Xre = ws + X_RE;
  float* Xim = ws + X_IM;
  for (int j = wv; j < 18; j += 8) {
    int comp = j % 2;
    int tile = j / 2;
    int kt = tile / 3, lt = tile % 3;
    const float* A1row = tabC + (kt * 16 + r) * 128;
    const float* A2row = tabS + (kt * 16 + r) * 128;
    v8f acc = {};
    for (int k0 = 0; k0 < 128; k0 += 4) {
      int kk = k0 + 2 * half;
      // re: C@Tre + S@Tim ;  im: C@Tim - S@Tre
      v2f a1; a1.x = A1row[kk]; a1.y = A1row[kk + 1];
      const float* B1 = comp ? sTim : sTre;
      v2f b1; b1.x = B1[kk * 48 + lt * 16 + r]; b1.y = B1[(kk + 1) * 48 + lt * 16 + r];
      acc = wmma4(a1, b1, acc);
      float sgn = comp ? -1.f : 1.f;
      v2f a2; a2.x = sgn * A2row[kk]; a2.y = sgn * A2row[kk + 1];
      const float* B2 = comp ? sTre : sTim;
      v2f b2; b2.x = B2[kk * 48 + lt * 16 + r]; b2.y = B2[(kk + 1) * 48 + lt * 16 + r];
      acc = wmma4(a2, b2, acc);
    }
    float* dst = comp ? Xim : Xre;
#pragma unroll
    for (int i = 0; i < 8; ++i) {
      int k = kt * 16 + i + 8 * half;
      int l = lt * 16 + r;
      if (k < 33 && l < 33)
        dst[(size_t)(k * 33 + l) * 1024 + bid] = acc[i];
    }
  }
}

// ---------------- kernel 3: per-mode complex channel mix  Y = X * R ----------------
__global__ void k_mix(float* __restrict__ ws) {
  __shared__ float sXre[1024], sXim[1024];
  __shared__ float sRre[4096], sRim[4096];
  const int tid = threadIdx.x;
  const int m = blockIdx.x;  // 0..1088
  const float* Xre = ws + X_RE + (size_t)m * 1024;
  const float* Xim = ws + X_IM + (size_t)m * 1024;
  const float* Rre = ws + RT_RE + (size_t)m * 4096;
  const float* Rim = ws + RT_IM + (size_t)m * 4096;

  // async stage operands into LDS (all coalesced b128)
  async_ld_b128(lds_off(&sXre[tid * 4]), Xre + tid * 4);
  async_ld_b128(lds_off(&sXim[tid * 4]), Xim + tid * 4);
#pragma unroll
  for (int i = 0; i < 4; ++i) {
    int e = tid + i * 256;  // float4 index within 4096 floats
    async_ld_b128(lds_off(&sRre[e * 4]), Rre + e * 4);
    async_ld_b128(lds_off(&sRim[e * 4]), Rim + e * 4);
  }
  wait_async0();
  __syncthreads();

  const int wv = __builtin_amdgcn_readfirstlane(tid >> 5);
  const int lane = tid & 31, half = lane >> 4, r = lane & 15;
  int comp = wv & 1;     // 0: Yre, 1: Yim
  int nt = wv >> 1;      // output o tile 0..3
  v8f acc = {};
  for (int k0 = 0; k0 < 64; k0 += 4) {
    int kk = k0 + 2 * half;
    // re: Xre@Rre - Xim@Rim ;  im: Xre@Rim + Xim@Rre
    v2f a1; a1.x = sXre[r * 64 + kk]; a1.y = sXre[r * 64 + kk + 1];
    const float* B1 = comp ? sRim : sRre;
    v2f b1; b1.x = B1[kk * 64 + nt * 16 + r]; b1.y = B1[(kk + 1) * 64 + nt * 16 + r];
    acc = wmma4(a1, b1, acc);
    float sgn = comp ? 1.f : -1.f;
    v2f a2; a2.x = sgn * sXim[r * 64 + kk]; a2.y = sgn * sXim[r * 64 + kk + 1];
    const float* B2 = comp ? sRre : sRim;
    v2f b2; b2.x = B2[kk * 64 + nt * 16 + r]; b2.y = B2[(kk + 1) * 64 + nt * 16 + r];
    acc = wmma4(a2, b2, acc);
  }
  float* dst = ws + (comp ? Y_IM : Y_RE) + (size_t)m * 1024;
#pragma unroll
  for (int i = 0; i < 8; ++i)
    dst[(i + 8 * half) * 64 + nt * 16 + r] = acc[i];
}

// ---------------- kernel 4: inverse truncated 2D DFT, one (b,o) image per block ----------------
__global__ void k_inv(const float* __restrict__ ws, float* __restrict__ out) {
  __shared__ float sYre[36 * 48], sYim[36 * 48];
  __shared__ float sUre[128 * 48], sUim[128 * 48];
  const int tid = threadIdx.x;
  const int wv = __builtin_amdgcn_readfirstlane(tid >> 5);
  const int lane = tid & 31;
  const int half = lane >> 4, r = lane & 15;
  const int bid = blockIdx.x;  // b*64+o

  for (int e = tid; e < 36 * 48; e += 256) { sYre[e] = 0.f; sYim[e] = 0.f; }
  __syncthreads();
  // per-lane-address async gather of the spectral tile into LDS
  const float* Yre = ws + Y_RE;
  const float* Yim = ws + Y_IM;
  for (int e = tid; e < 1089; e += 256) {
    int k = e / 33, l = e % 33;
    async_ld_b32(lds_off(&sYre[k * 48 + l]), Yre + (size_t)e * 1024 + bid);
    async_ld_b32(lds_off(&sYim[k * 48 + l]), Yim + (size_t)e * 1024 + bid);
  }
  wait_async0();
  __syncthreads();

  const float* tabCt = ws + TAB_CT;
  const float* tabSt = ws + TAB_ST;
  // U: M=128 h (8 tiles), K=36 k (zero-padded), N=48 l (3 tiles), x2 comp
  for (int j = wv; j < 48; j += 8) {
    int comp = j & 1;
    int tile = j >> 1;
    int ht = tile / 3, lt = tile % 3;
    const float* A1row = tabCt + (ht * 16 + r) * 48;
    const float* A2row = tabSt + (ht * 16 + r) * 48;
    v8f acc = {};
    for (int k0 = 0; k0 < 36; k0 += 4) {
      int kk = k0 + 2 * half;
      // re: Ct@Yre - St@Yim ;  im: Ct@Yim + St@Yre
      v2f a1; a1.x = A1row[kk]; a1.y = A1row[kk + 1];
      const float* B1 = comp ? sYim : sYre;
      v2f b1; b1.x = B1[kk * 48 + lt * 16 + r]; b1.y = B1[(kk + 1) * 48 + lt * 16 + r];
      acc = wmma4(a1, b1, acc);
      float sgn = comp ? 1.f : -1.f;
      v2f a2; a2.x = sgn * A2row[kk]; a2.y = sgn * A2row[kk + 1];
      const float* B2 = comp ? sYre : sYim;
      v2f b2; b2.x = B2[kk * 48 + lt * 16 + r]; b2.y = B2[(kk + 1) * 48 + lt * 16 + r];
      acc = wmma4(a2, b2, acc);
    }
    float* dstU = comp ? sUim : sUre;
#pragma unroll
    for (int i = 0; i < 8; ++i)
      dstU[(ht * 16 + i + 8 * half) * 48 + lt * 16 + r] = acc[i];
  }
  __syncthreads();

  const float* tabC = ws + TAB_C;
  const float* tabS = ws + TAB_S;
  float* ob = out + (size_t)bid * 16384;
  // out: M=128 h (8 tiles), K=36 l (zero-padded), N=128 w (8 tiles)
  for (int j = wv; j < 64; j += 8) {
    int ht = j >> 3, wt = j & 7;
    const float* A1row = sUre + (ht * 16 + r) * 48;
    const float* A2row = sUim + (ht * 16 + r) * 48;
    v8f acc = {};
    for (int k0 = 0; k0 < 36; k0 += 4) {
      int kk = k0 + 2 * half;
      v2f a1; a1.x = A1row[kk]; a1.y = A1row[kk + 1];
      v2f b1; b1.x = tabC[kk * 128 + wt * 16 + r]; b1.y = tabC[(kk + 1) * 128 + wt * 16 + r];
      acc = wmma4(a1, b1, acc);
      v2f a2; a2.x = -A2row[kk]; a2.y = -A2row[kk + 1];
      v2f b2; b2.x = tabS[kk * 128 + wt * 16 + r]; b2.y = tabS[(kk + 1) * 128 + wt * 16 + r];
      acc = wmma4(a2, b2, acc);
    }
#pragma unroll
    for (int i = 0; i < 8; ++i)
      ob[(ht * 16 + i + 8 * half) * 128 + wt * 16 + r] = acc[i] * (1.0f / 16384.0f);
  }
}

// ---------------- launcher ----------------
extern "C" void kernel_launch(void* const* d_in, const int* in_sizes, int n_in,
                              void* d_out, int out_size, void* d_ws, size_t ws_size,
                              hipStream_t stream) {
  (void)in_sizes; (void)n_in; (void)out_size; (void)ws_size;
  const float* x  = (const float*)d_in[0];
  const float* wr = (const float*)d_in[1];
  const float* wi = (const float*)d_in[2];
  float* out = (float*)d_out;
  float* ws  = (float*)d_ws;

  k_tables<<<24, 256, 0, stream>>>(ws);
  k_transpose<<<dim3(35, 128, 2), dim3(32, 8), 0, stream>>>(wr, wi, ws);
  k_fwd<<<1024, 256, 0, stream>>>(x, ws);
  k_mix<<<1089, 256, 0, stream>>>(ws);
  k_inv<<<1024, 256, 0, stream>>>(ws, out);
}